// EdgeGATLayer_59236188946482
// MI455X (gfx1250) — compile-verified
//
#include <hip/hip_runtime.h>
#include <hip/hip_bf16.h>

typedef float v2f __attribute__((ext_vector_type(2)));
typedef float v8f __attribute__((ext_vector_type(8)));

#define IN_NODE 256
#define IN_EDGE 64
#define OUT_DIM 128

// float atomic max via sign-aware int/uint ordering trick.
// Works with initial bit pattern 0xFFFFFFFF (memset 0xFF): int view is -1
// (below every non-negative float pattern), uint view is UINT_MAX (above
// every negative float pattern).
__device__ __forceinline__ void atomicMaxF(float* addr, float val) {
    if (__float_as_int(val) >= 0) {
        atomicMax((int*)addr, __float_as_int(val));
    } else {
        atomicMin((unsigned int*)addr, __float_as_uint(val));
    }
}

// ---------------------------------------------------------------------------
// Kernel 1: z = feats_node @ Wn^T  (wave computes a 16x128 tile via f32 WMMA)
// Fused: s1[n] = dot(z[n], Wa[0:128]), s2[n] = dot(z[n], Wa[128:256])
// A layout (16x4 f32): lanes 0-15 row=lane K={k,k+1}; lanes 16-31 row=lane-16
// K={k+2,k+3}.  B (4x16): mirrored -> each lane holds B[k+koff..+1][nlane].
// C/D (16x16 f32): VGPR r -> (M = r + 8*half, N = nlane).
// ---------------------------------------------------------------------------
__global__ void node_proj_wmma(const float* __restrict__ X,
                               const float* __restrict__ Wn,
                               const float* __restrict__ Wa,
                               float* __restrict__ z,
                               float* __restrict__ s1,
                               float* __restrict__ s2,
                               int n_nodes) {
    const int lane  = threadIdx.x & 31;
    const int wave  = (blockIdx.x * blockDim.x + threadIdx.x) >> 5;
    const int ntile = (n_nodes + 15) >> 4;
    if (wave >= ntile) return;                 // wave-uniform: EXEC stays all-1s
    const int m0    = wave << 4;
    const int nlane = lane & 15;
    const int half  = lane >> 4;               // 0 or 1
    const int koff  = half * 2;

    const float* __restrict__ arow = X + (size_t)(m0 + nlane) * IN_NODE;

    float p1[8], p2[8];
#pragma unroll
    for (int r = 0; r < 8; ++r) { p1[r] = 0.f; p2[r] = 0.f; }

#pragma unroll
    for (int t = 0; t < 8; ++t) {              // 8 tiles of 16 output columns
        const int n = t * 16 + nlane;
        const float* __restrict__ brow = Wn + (size_t)n * IN_NODE;
        v8f acc = {};
#pragma unroll 4
        for (int k = 0; k < IN_NODE; k += 4) {
            v2f a = *(const v2f*)(arow + k + koff);
            v2f b = *(const v2f*)(brow + k + koff);
            acc = __builtin_amdgcn_wmma_f32_16x16x4_f32(
                false, a, false, b, (short)0, acc, false, false);
        }
        const float wa1 = Wa[n];
        const float wa2 = Wa[OUT_DIM + n];
#pragma unroll
        for (int r = 0; r < 8; ++r) {
            const int mrow = m0 + r + half * 8;
            z[(size_t)mrow * OUT_DIM + n] = acc[r];
            p1[r] += acc[r] * wa1;
            p2[r] += acc[r] * wa2;
        }
    }
    // Reduce the attention partials over the 16 lanes of each half-wave.
#pragma unroll
    for (int r = 0; r < 8; ++r) {
#pragma unroll
        for (int off = 1; off < 16; off <<= 1) {
            p1[r] += __shfl_xor(p1[r], off, 32);
            p2[r] += __shfl_xor(p2[r], off, 32);
        }
    }
    if (nlane == 0) {
#pragma unroll
        for (int r = 0; r < 8; ++r) {
            const int mrow = m0 + r + half * 8;
            s1[mrow] = p1[r];
            s2[mrow] = p2[r];
        }
    }
}

// ---------------------------------------------------------------------------
// Kernel 2: ze = feats_edge @ We^T  (16x64 tile per wave, We staged in LDS)
// Fused: zeDot[e] = dot(ze[e], Wa[256:320])
// ---------------------------------------------------------------------------
__global__ void edge_proj_wmma(const float* __restrict__ Xe,
                               const float* __restrict__ We,
                               const float* __restrict__ Wa,
                               float* __restrict__ ze,
                               float* __restrict__ zeDot,
                               int n_edges) {
    __shared__ float lwe[IN_EDGE * 66];        // padded stride to dodge bank conflicts
    for (int i = threadIdx.x; i < IN_EDGE * IN_EDGE; i += blockDim.x) {
        lwe[(i >> 6) * 66 + (i & 63)] = We[i];
    }
    __syncthreads();

    const int lane  = threadIdx.x & 31;
    const int wave  = (blockIdx.x * blockDim.x + threadIdx.x) >> 5;
    const int ntile = n_edges >> 4;
    if (wave >= ntile) return;
    const int m0    = wave << 4;
    const int nlane = lane & 15;
    const int half  = lane >> 4;
    const int koff  = half * 2;

    const float* __restrict__ arow = Xe + (size_t)(m0 + nlane) * IN_EDGE;
    const float* __restrict__ WaE  = Wa + 2 * OUT_DIM;

    float pe[8];
#pragma unroll
    for (int r = 0; r < 8; ++r) pe[r] = 0.f;

#pragma unroll
    for (int t = 0; t < 4; ++t) {              // 4 tiles of 16 output columns
        const int n = t * 16 + nlane;
        v8f acc = {};
#pragma unroll
        for (int k = 0; k < IN_EDGE; k += 4) {
            v2f a = *(const v2f*)(arow + k + koff);
            v2f b = *(const v2f*)(&lwe[n * 66 + k + koff]);
            acc = __builtin_amdgcn_wmma_f32_16x16x4_f32(
                false, a, false, b, (short)0, acc, false, false);
        }
        const float wae = WaE[n];
#pragma unroll
        for (int r = 0; r < 8; ++r) {
            const int mrow = m0 + r + half * 8;
            ze[(size_t)mrow * IN_EDGE + n] = acc[r];
            pe[r] += acc[r] * wae;
        }
    }
#pragma unroll
    for (int r = 0; r < 8; ++r) {
#pragma unroll
        for (int off = 1; off < 16; off <<= 1) {
            pe[r] += __shfl_xor(pe[r], off, 32);
        }
    }
    if (nlane == 0) {
#pragma unroll
        for (int r = 0; r < 8; ++r) {
            zeDot[m0 + r + half * 8] = pe[r];
        }
    }
}

// ---------------------------------------------------------------------------
// Kernel 3: per-edge logit + leaky relu + segment max (atomic)
// ---------------------------------------------------------------------------
__global__ void edge_logit(const int* __restrict__ src, const int* __restrict__ dst,
                           const float* __restrict__ s1, const float* __restrict__ s2,
                           const float* __restrict__ zeDot,
                           float* __restrict__ elog, float* __restrict__ mmax,
                           int n_edges) {
    int i = blockIdx.x * blockDim.x + threadIdx.x;
    if (i >= n_edges) return;
    const int d = dst[i];
    float v = s1[src[i]] + s2[d] + zeDot[i];
    float e = v > 0.f ? v : 0.2f * v;
    elog[i] = e;
    atomicMaxF(&mmax[d], e);
}

// ---------------------------------------------------------------------------
// Kernel 4: e_exp = exp(e - m[dst]); denom = segment_sum(e_exp)
// ---------------------------------------------------------------------------
__global__ void edge_exp(const int* __restrict__ dst,
                         const float* __restrict__ elog,
                         const float* __restrict__ mmax,
                         float* __restrict__ eexp_, float* __restrict__ denom,
                         int n_edges) {
    int i = blockIdx.x * blockDim.x + threadIdx.x;
    if (i >= n_edges) return;
    const int d = dst[i];
    float ee = expf(elog[i] - mmax[d]);
    eexp_[i] = ee;
    atomicAdd(&denom[d], ee);
}

// ---------------------------------------------------------------------------
// Kernel 5: h[dst] += alpha * z[src]   (one wave per edge, coalesced atomics;
// z and h are L2-resident so this stays on-chip)
// ---------------------------------------------------------------------------
__global__ void scatter_h(const int* __restrict__ src, const int* __restrict__ dst,
                          const float* __restrict__ eexp_, const float* __restrict__ denom,
                          const float* __restrict__ z, float* __restrict__ h,
                          int n_edges) {
    const int lane = threadIdx.x & 31;
    const int edge = (blockIdx.x * blockDim.x + threadIdx.x) >> 5;
    if (edge >= n_edges) return;
    const int s = src[edge];
    const int d = dst[edge];
    const float alpha = eexp_[edge] / fmaxf(denom[d], 1e-38f);
    const float* __restrict__ zs = z + (size_t)s * OUT_DIM;
    float* __restrict__ hd = h + (size_t)d * OUT_DIM;
#pragma unroll
    for (int f = lane; f < OUT_DIM; f += 32) {
        atomicAdd(&hd[f], alpha * zs[f]);
    }
}

// ---------------------------------------------------------------------------
extern "C" void kernel_launch(void* const* d_in, const int* in_sizes, int n_in,
                              void* d_out, int out_size, void* d_ws, size_t ws_size,
                              hipStream_t stream) {
    const float* feats_node = (const float*)d_in[0];
    const float* feats_edge = (const float*)d_in[1];
    const int*   src        = (const int*)d_in[2];
    const int*   dst        = (const int*)d_in[3];
    const float* Wn         = (const float*)d_in[4];
    const float* We         = (const float*)d_in[5];
    const float* Wa         = (const float*)d_in[6];

    const int n_nodes = in_sizes[0] / IN_NODE;
    const int n_edges = in_sizes[2];

    float* h  = (float*)d_out;                       // [n_nodes, 128]
    float* ze = h + (size_t)n_nodes * OUT_DIM;       // [n_edges, 64]

    // workspace carve-out (~36 MB)
    float* z     = (float*)d_ws;                     // [n_nodes, 128]
    float* s1    = z + (size_t)n_nodes * OUT_DIM;    // [n_nodes]
    float* s2    = s1 + n_nodes;                     // [n_nodes]
    float* mmax  = s2 + n_nodes;                     // [n_nodes]
    float* denom = mmax + n_nodes;                   // [n_nodes]
    float* zeDot = denom + n_nodes;                  // [n_edges]
    float* elog  = zeDot + n_edges;                  // [n_edges]
    float* eexp_ = elog + n_edges;                   // [n_edges]

    hipMemsetAsync(h,     0,    (size_t)n_nodes * OUT_DIM * sizeof(float), stream);
    hipMemsetAsync(mmax,  0xFF, (size_t)n_nodes * sizeof(float), stream); // -inf under trick
    hipMemsetAsync(denom, 0,    (size_t)n_nodes * sizeof(float), stream);

    {   // node projection: one wave per 16 nodes
        int ntiles = (n_nodes + 15) / 16;
        int blocks = (ntiles * 32 + 255) / 256;
        node_proj_wmma<<<blocks, 256, 0, stream>>>(feats_node, Wn, Wa, z, s1, s2, n_nodes);
    }
    {   // edge projection: one wave per 16 edges
        int ntiles = n_edges / 16;
        int blocks = (ntiles * 32 + 255) / 256;
        edge_proj_wmma<<<blocks, 256, 0, stream>>>(feats_edge, We, Wa, ze, zeDot, n_edges);
    }
    {
        int blocks = (n_edges + 255) / 256;
        edge_logit<<<blocks, 256, 0, stream>>>(src, dst, s1, s2, zeDot, elog, mmax, n_edges);
        edge_exp<<<blocks, 256, 0, stream>>>(dst, elog, mmax, eexp_, denom, n_edges);
    }
    {   // weighted scatter: one wave per edge
        long long waves = (long long)n_edges;
        int blocks = (int)((waves * 32 + 255) / 256);
        scatter_h<<<blocks, 256, 0, stream>>>(src, dst, eexp_, denom, z, h, n_edges);
    }
}